// test_model_24696061952013
// MI455X (gfx1250) — compile-verified
//
#include <hip/hip_runtime.h>
#include <math.h>

typedef __attribute__((ext_vector_type(2))) float v2f;
typedef __attribute__((ext_vector_type(8))) float v8f;

#define EPS 1e-3f

// ---------------------------------------------------------------------------
// channel mean over H*W for [B,HW,C] layout -> dst[b*C+c]
// ---------------------------------------------------------------------------
__global__ void channel_mean_kernel(const float* __restrict__ src,
                                    float* __restrict__ dst,
                                    int HW, int C) {
    int bc = blockIdx.x;
    int b = bc / C, c = bc % C;
    __shared__ float red[256];
    float s = 0.f;
    const float* p = src + (size_t)b * HW * C + c;
    for (int i = threadIdx.x; i < HW; i += 256) s += p[(size_t)i * C];
    red[threadIdx.x] = s;
    __syncthreads();
    for (int off = 128; off > 0; off >>= 1) {
        if (threadIdx.x < off) red[threadIdx.x] += red[threadIdx.x + off];
        __syncthreads();
    }
    if (threadIdx.x == 0) dst[b * C + c] = red[0] / (float)HW;
}

// ---------------------------------------------------------------------------
// route = sigmoid(pooled @ rw + rb); mk[b] = se[ci] * sum_e route[b,e]*experts[e]
// The SE gate (scale of input channel ci) is folded into the mixed kernel:
// conv(x*s, K) == conv(x, K with rows ci scaled by s[ci]).
// experts flat: [3, K*N]; element idx -> k = idx/COUT, ci = k % CIN.
// ---------------------------------------------------------------------------
__global__ void route_mix_kernel(const float* __restrict__ pooled,
                                 const float* __restrict__ rw,
                                 const float* __restrict__ rb,
                                 const float* __restrict__ experts,
                                 const float* __restrict__ se,   // may be null
                                 float* __restrict__ mk,
                                 int Cin, int Cout, int KN) {
    int b = blockIdx.x;
    float r[3];
#pragma unroll
    for (int e = 0; e < 3; ++e) {
        float a = rb[e];
        for (int c = 0; c < Cin; ++c) a += pooled[b * Cin + c] * rw[c * 3 + e];
        r[e] = 1.f / (1.f + expf(-a));
    }
    int idx = blockIdx.y * blockDim.x + threadIdx.x;
    if (idx < KN) {
        float v = r[0] * experts[idx] + r[1] * experts[(size_t)KN + idx] +
                  r[2] * experts[2 * (size_t)KN + idx];
        if (se) {
            int ci = (idx / Cout) % Cin;
            v *= se[b * Cin + ci];
        }
        mk[(size_t)b * KN + idx] = v;
    }
}

// ---------------------------------------------------------------------------
// 2x2 maxpool, stride 2, NHWC
// ---------------------------------------------------------------------------
__global__ void maxpool2_kernel(const float* __restrict__ src,
                                float* __restrict__ dst,
                                int Hp, int Wp, int C, int total) {
    int i = blockIdx.x * blockDim.x + threadIdx.x;
    if (i >= total) return;
    int c = i % C;
    int t = i / C;
    int x = t % Wp; t /= Wp;
    int y = t % Hp;
    int b = t / Hp;
    int W = Wp * 2;
    const float* p = src + (((size_t)b * (Hp * 2) + 2 * y) * W + 2 * x) * C + c;
    size_t rs = (size_t)W * C;
    float v = fmaxf(fmaxf(p[0], p[C]), fmaxf(p[rs], p[rs + C]));
    dst[i] = v;
}

// ---------------------------------------------------------------------------
// SE gate: s = sigmoid(relu(mean @ w1) @ w2); also pooled_next = mean * s
// (pooled_next feeds the next layer's router; activation scaling is folded
//  into the next layer's mixed kernel by route_mix_kernel)
// ---------------------------------------------------------------------------
__global__ void se_kernel(const float* __restrict__ meanb,
                          const float* __restrict__ w1,
                          const float* __restrict__ w2,
                          float* __restrict__ s_out,
                          float* __restrict__ pooled_next,
                          int C, int Hd) {
    int b = blockIdx.x;
    __shared__ float hid[8];
    int t = threadIdx.x;
    if (t < Hd) {
        float a = 0.f;
        for (int c = 0; c < C; ++c) a += meanb[b * C + c] * w1[c * Hd + t];
        hid[t] = fmaxf(a, 0.f);
    }
    __syncthreads();
    if (t < C) {
        float a = 0.f;
        for (int j = 0; j < Hd; ++j) a += hid[j] * w2[j * C + t];
        float s = 1.f / (1.f + expf(-a));
        s_out[b * C + t] = s;
        pooled_next[b * C + t] = meanb[b * C + t] * s;
    }
}

// ---------------------------------------------------------------------------
// CondConv as per-sample implicit GEMM on V_WMMA_F32_16X16X4_F32.
// One wave computes a 32x16 output tile (two 16x16 accumulators sharing the
// B fragment). Loop structure: 9 taps outer (index math + validity hoisted),
// contiguous input channels inner (aligned 8-byte vector A loads).
// Fragment layouts per ISA 7.12.2 (f32, wave32):
//   A: lanes 0-15 -> M=lane, {K=k0,k0+1}; lanes 16-31 -> M=lane-16, {K=k0+2,k0+3}
//   B: same striping with N in place of M
//   C/D: VGPR v -> row v (lanes 0-15) / v+8 (lanes 16-31), col = lane&15
// BN(+ReLU) fused in epilogue. MEANOUT mode reduces the tile and atomically
// accumulates raw conv sums (BN applied in finalize; mean∘affine==affine∘mean).
// ---------------------------------------------------------------------------
template <int H, int W, int CIN, int COUT, bool RELU, bool MEANOUT>
__global__ __launch_bounds__(32) void condconv_wmma_kernel(
    const float* __restrict__ in, const float* __restrict__ mk,
    const float* __restrict__ bn_g, const float* __restrict__ bn_b,
    const float* __restrict__ bn_m, const float* __restrict__ bn_v,
    float* __restrict__ out) {
    constexpr int K = 9 * CIN;
    const int b    = blockIdx.z;
    const int tm   = blockIdx.x;   // 32-row M tile
    const int tn   = blockIdx.y;   // 16-col N tile
    const int lane = threadIdx.x;
    const int half = lane >> 4;
    const int l15  = lane & 15;

    const int m0  = tm * 32 + l15;       // A-side pixel rows for the two tiles
    const int m1  = m0 + 16;
    const int py0 = m0 / W, px0 = m0 % W;
    const int py1 = m1 / W, px1 = m1 % W;
    const int n   = tn * 16 + l15;       // B-side output channel

    const float* __restrict__ inb = in + (size_t)b * H * W * CIN + 2 * half;
    const float* __restrict__ mkb =
        mk + (size_t)b * K * COUT + (size_t)(2 * half) * COUT + n;

    v8f acc0 = {}, acc1 = {};
    for (int tap = 0; tap < 9; ++tap) {
        const int ky  = tap / 3, kx = tap % 3;
        const int iy0 = py0 + ky - 1, ix0 = px0 + kx - 1;
        const int iy1 = py1 + ky - 1, ix1 = px1 + kx - 1;
        const bool ok0 = (unsigned)iy0 < (unsigned)H && (unsigned)ix0 < (unsigned)W;
        const bool ok1 = (unsigned)iy1 < (unsigned)H && (unsigned)ix1 < (unsigned)W;
        const float* a0p = inb + ((size_t)iy0 * W + ix0) * CIN;
        const float* a1p = inb + ((size_t)iy1 * W + ix1) * CIN;
        const float* bp  = mkb + (size_t)tap * CIN * COUT;
        for (int c0 = 0; c0 < CIN; c0 += 4) {
            // per-lane-uniform K-tail guard; compiled out when CIN % 4 == 0
            const bool kin = (CIN % 4 == 0) || (c0 + 2 * half + 1 < CIN);
            v2f a0 = {}, a1 = {}, bf = {};
            if (ok0 && kin) a0 = *(const v2f*)(a0p + c0);   // 8B aligned: CIN even
            if (ok1 && kin) a1 = *(const v2f*)(a1p + c0);
            if (kin) {
                bf[0] = bp[(size_t)c0 * COUT];
                bf[1] = bp[(size_t)c0 * COUT + COUT];
            }
            acc0 = __builtin_amdgcn_wmma_f32_16x16x4_f32(
                false, a0, false, bf, (short)0, acc0, false, false);
            acc1 = __builtin_amdgcn_wmma_f32_16x16x4_f32(
                false, a1, false, bf, (short)0, acc1, false, false);
        }
    }

    if (MEANOUT) {
        float s = acc0[0] + acc0[1] + acc0[2] + acc0[3] +
                  acc0[4] + acc0[5] + acc0[6] + acc0[7] +
                  acc1[0] + acc1[1] + acc1[2] + acc1[3] +
                  acc1[4] + acc1[5] + acc1[6] + acc1[7];
        s += __shfl_xor(s, 16, 32);     // combine lanes l and l+16 (same n)
        if (half == 0) atomicAdd(&out[(size_t)b * COUT + n], s);
    } else {
        float inv   = bn_g[n] / sqrtf(bn_v[n] + EPS);
        float shift = bn_b[n] - bn_m[n] * inv;
        float* outb = out + (size_t)b * H * W * COUT + n;
#pragma unroll
        for (int v = 0; v < 8; ++v) {
            int mm = tm * 32 + v + 8 * half;
            float v0 = acc0[v] * inv + shift;
            float v1 = acc1[v] * inv + shift;
            if (RELU) { v0 = fmaxf(v0, 0.f); v1 = fmaxf(v1, 0.f); }
            outb[(size_t)mm * COUT]        = v0;
            outb[(size_t)(mm + 16) * COUT] = v1;
        }
    }
}

__global__ void zero_kernel(float* __restrict__ p, int n) {
    int i = blockIdx.x * blockDim.x + threadIdx.x;
    if (i < n) p[i] = 0.f;
}

__global__ void finalize_bn_kernel(const float* __restrict__ acc,
                                   const float* __restrict__ g,
                                   const float* __restrict__ bta,
                                   const float* __restrict__ mu,
                                   const float* __restrict__ var,
                                   float* __restrict__ out,
                                   int total, int C, float invHW) {
    int i = blockIdx.x * blockDim.x + threadIdx.x;
    if (i >= total) return;
    int n = i % C;
    float inv = g[n] / sqrtf(var[n] + EPS);
    out[i] = acc[i] * invHW * inv + (bta[n] - mu[n] * inv);
}

// ---------------------------------------------------------------------------
extern "C" void kernel_launch(void* const* d_in, const int* in_sizes, int n_in,
                              void* d_out, int out_size, void* d_ws, size_t ws_size,
                              hipStream_t stream) {
    const int B = 64;
    const float* x        = (const float*)d_in[0];
    const float* experts1 = (const float*)d_in[1];
    const float* rw1      = (const float*)d_in[2];
    const float* rb1      = (const float*)d_in[3];
    const float* bn1g = (const float*)d_in[4];  const float* bn1b = (const float*)d_in[5];
    const float* bn1m = (const float*)d_in[6];  const float* bn1v = (const float*)d_in[7];
    const float* se1w1 = (const float*)d_in[8]; const float* se1w2 = (const float*)d_in[9];
    const float* experts2 = (const float*)d_in[10];
    const float* rw2      = (const float*)d_in[11];
    const float* rb2      = (const float*)d_in[12];
    const float* bn2g = (const float*)d_in[13]; const float* bn2b = (const float*)d_in[14];
    const float* bn2m = (const float*)d_in[15]; const float* bn2v = (const float*)d_in[16];
    const float* se2w1 = (const float*)d_in[17]; const float* se2w2 = (const float*)d_in[18];
    const float* experts3 = (const float*)d_in[19];
    const float* rw3      = (const float*)d_in[20];
    const float* rb3      = (const float*)d_in[21];
    const float* bn3g = (const float*)d_in[22]; const float* bn3b = (const float*)d_in[23];
    const float* bn3m = (const float*)d_in[24]; const float* bn3v = (const float*)d_in[25];
    float* out = (float*)d_out;

    // workspace layout (floats)
    float* ws = (float*)d_ws;
    size_t off = 0;
    float* hbuf   = ws + off; off += (size_t)B * 128 * 128 * 32;  // h1, reused for h2
    float* p1     = ws + off; off += (size_t)B * 64 * 64 * 32;
    float* p2     = ws + off; off += (size_t)B * 32 * 32 * 64;
    float* mkbuf  = ws + off; off += (size_t)B * 576 * 128;       // mk1/mk2/mk3 (reused)
    float* pooled = ws + off; off += (size_t)B * 128;             // router input
    float* meanb  = ws + off; off += (size_t)B * 128;             // SE channel means
    float* sbuf   = ws + off; off += (size_t)B * 128;             // SE scales (s1 then s2)
    float* accum  = ws + off; off += (size_t)B * 128;             // layer-3 sums
    float* h2 = hbuf;

    // ---- stage 1: route1 -> mixed kernels -> conv1 (+BN+ReLU) ----
    channel_mean_kernel<<<B * 6, 256, 0, stream>>>(x, pooled, 128 * 128, 6);
    {
        int KN = 9 * 6 * 32;
        route_mix_kernel<<<dim3(B, (KN + 255) / 256), 256, 0, stream>>>(
            pooled, rw1, rb1, experts1, nullptr, mkbuf, 6, 32, KN);
    }
    condconv_wmma_kernel<128, 128, 6, 32, true, false>
        <<<dim3(128 * 128 / 32, 32 / 16, B), 32, 0, stream>>>(
            x, mkbuf, bn1g, bn1b, bn1m, bn1v, hbuf);

    // ---- maxpool1 + SE1 ----
    {
        int total = B * 64 * 64 * 32;
        maxpool2_kernel<<<(total + 255) / 256, 256, 0, stream>>>(hbuf, p1, 64, 64, 32, total);
    }
    channel_mean_kernel<<<B * 32, 256, 0, stream>>>(p1, meanb, 64 * 64, 32);
    se_kernel<<<B, 64, 0, stream>>>(meanb, se1w1, se1w2, sbuf, pooled, 32, 2);

    // ---- stage 2: route2 (SE1 folded into mixed kernel) -> conv2 ----
    {
        int KN = 9 * 32 * 64;
        route_mix_kernel<<<dim3(B, (KN + 255) / 256), 256, 0, stream>>>(
            pooled, rw2, rb2, experts2, sbuf, mkbuf, 32, 64, KN);
    }
    condconv_wmma_kernel<64, 64, 32, 64, true, false>
        <<<dim3(64 * 64 / 32, 64 / 16, B), 32, 0, stream>>>(
            p1, mkbuf, bn2g, bn2b, bn2m, bn2v, h2);

    // ---- maxpool2 + SE2 ----
    {
        int total = B * 32 * 32 * 64;
        maxpool2_kernel<<<(total + 255) / 256, 256, 0, stream>>>(h2, p2, 32, 32, 64, total);
    }
    channel_mean_kernel<<<B * 64, 256, 0, stream>>>(p2, meanb, 32 * 32, 64);
    se_kernel<<<B, 64, 0, stream>>>(meanb, se2w1, se2w2, sbuf, pooled, 64, 4);

    // ---- stage 3: route3 (SE2 folded) -> conv3 with fused spatial mean ----
    {
        int KN = 9 * 64 * 128;
        route_mix_kernel<<<dim3(B, (KN + 255) / 256), 256, 0, stream>>>(
            pooled, rw3, rb3, experts3, sbuf, mkbuf, 64, 128, KN);
    }
    zero_kernel<<<(B * 128 + 255) / 256, 256, 0, stream>>>(accum, B * 128);
    condconv_wmma_kernel<32, 32, 64, 128, false, true>
        <<<dim3(32 * 32 / 32, 128 / 16, B), 32, 0, stream>>>(
            p2, mkbuf, bn3g, bn3b, bn3m, bn3v, accum);
    finalize_bn_kernel<<<(B * 128 + 255) / 256, 256, 0, stream>>>(
        accum, bn3g, bn3b, bn3m, bn3v, out, B * 128, 128, 1.f / (32.f * 32.f));
}